// PoolingLayer_69286412419080
// MI455X (gfx1250) — compile-verified
//
#include <hip/hip_runtime.h>
#include <stdint.h>

// Segment-mean + ReLU over contiguous row-segments of a [N, 512] f32 matrix.
// MI455X (gfx1250) strategy: HBM-bandwidth-bound streaming reduction using the
// CDNA5 async HBM->LDS DMA path (GLOBAL_LOAD_ASYNC_TO_LDS_B128 / ASYNCcnt),
// double-buffered 16-row tiles, no workgroup barriers (each thread consumes
// exactly the bytes it DMA'd).

#define D_FEAT   512
#define TILE     16          // rows staged per LDS buffer
#define THREADS  128         // 4 waves; thread t owns columns [4t, 4t+4)

__device__ __forceinline__ void async_row_b128(unsigned lds_byte_off,
                                               unsigned g_byte_off,
                                               const float* base) {
    // GVS mode: copies 16 bytes from global address base (SGPR) + g_byte_off (VGPR) to LDS address lds_byte_off
    asm volatile("global_load_async_to_lds_b128 %0, %1, %2"
                 :: "v"(lds_byte_off), "v"(g_byte_off), "s"(base)
                 : "memory");
}
__device__ __forceinline__ void wait_asynccnt_le16() {
    asm volatile("s_wait_asynccnt 16" ::: "memory");
}
__device__ __forceinline__ void wait_dscnt_0() {
    asm volatile("s_wait_dscnt 0" ::: "memory");
}

__global__ __launch_bounds__(THREADS)
void seg_mean_relu_kernel(const float* __restrict__ x,
                          const int* __restrict__ idx,
                          float* __restrict__ out,
                          int num_nodes, int num_graphs)
{
    __shared__ float4 smem[2][TILE][THREADS];   // 64 KB

    const int g   = blockIdx.x;
    const int tid = threadIdx.x;

    const int r0 = idx[g];
    const int r1 = (g + 1 < num_graphs) ? idx[g + 1] : num_nodes;
    const int count = r1 - r0;                  // >= 1 (idx strictly increasing)
    const int nt = (count + TILE - 1) / TILE;

    // Low 32 bits of a generic LDS pointer == byte offset within LDS allocation.
    const unsigned lds_base  = (unsigned)(uintptr_t)(&smem[0][0][0]);
    const unsigned lane_byte = (unsigned)tid * 16u;

    // Issue one 16-row tile: 16 async b128 ops per wave, each wave/thread
    // loading exactly the 16B it will consume. Rows are clamped at the segment
    // end so every tile issues exactly TILE ops (keeps the wait immediate
    // constant); clamped slots are never read back.
    auto issue_tile = [&](int t, int buf) {
        const int tstart = r0 + t * TILE;
#pragma unroll
        for (int s = 0; s < TILE; ++s) {
            int r = tstart + s;
            if (r >= r1) r = r1 - 1;                       // pad/phantom load
            const unsigned goff = (unsigned)r * (D_FEAT * 4u) + lane_byte;
            const unsigned loff = lds_base
                + (unsigned)(buf * (TILE * THREADS * 16) + s * (THREADS * 16))
                + lane_byte;
            async_row_b128(loff, goff, x);
        }
    };

    float4 acc; acc.x = 0.f; acc.y = 0.f; acc.z = 0.f; acc.w = 0.f;

    issue_tile(0, 0);                            // prologue: tile 0 in flight
    for (int t = 0; t < nt; ++t) {
        const int buf = t & 1;
        // DS reads of buf^1 (iteration t-1) must retire before DMA overwrites it.
        wait_dscnt_0();
        issue_tile(t + 1, buf ^ 1);              // phantom past the end is clamped
        // 16 (tile t) + 16 (tile t+1) in flight; async loads complete in order,
        // so ASYNCcnt<=16 => tile t fully landed in LDS.
        wait_asynccnt_le16();

        const int tstart = r0 + t * TILE;
        int valid = r1 - tstart; if (valid > TILE) valid = TILE;
#pragma unroll 4
        for (int s = 0; s < valid; ++s) {
            float4 v = smem[buf][s][tid];        // ds_load_b128
            acc.x += v.x; acc.y += v.y; acc.z += v.z; acc.w += v.w;
        }
    }
    // Leftover phantom DMAs only touch the unread buffer; S_ENDPGM's implicit
    // wait-idle drains them.

    const float inv = 1.0f / (float)count;
    float4 o;
    o.x = fmaxf(acc.x * inv, 0.f);
    o.y = fmaxf(acc.y * inv, 0.f);
    o.z = fmaxf(acc.z * inv, 0.f);
    o.w = fmaxf(acc.w * inv, 0.f);
    reinterpret_cast<float4*>(out)[(size_t)g * (D_FEAT / 4) + tid] = o;
}

extern "C" void kernel_launch(void* const* d_in, const int* in_sizes, int n_in,
                              void* d_out, int out_size, void* d_ws, size_t ws_size,
                              hipStream_t stream) {
    (void)n_in; (void)out_size; (void)d_ws; (void)ws_size;
    const float* x  = (const float*)d_in[0];
    const int* idx  = (const int*)d_in[1];
    const int num_graphs = in_sizes[1];
    const int num_nodes  = in_sizes[0] / D_FEAT;
    seg_mean_relu_kernel<<<num_graphs, THREADS, 0, stream>>>(
        x, idx, (float*)d_out, num_nodes, num_graphs);
}